// GCN_Layer_42374147342489
// MI455X (gfx1250) — compile-verified
//
#include <hip/hip_runtime.h>

typedef __attribute__((ext_vector_type(2))) float v2f;
typedef __attribute__((ext_vector_type(8))) float v8f;
typedef __attribute__((ext_vector_type(4))) unsigned int u32x4;
typedef __attribute__((ext_vector_type(8))) int i32x8;
typedef __attribute__((ext_vector_type(4))) int i32x4;

#define N_NODES 100000
#define IN_DIM  256
#define HID_DIM 256
#define N_EDGES 3200000

// -------------------------------------------------------------------------
// TDM: DMA a 64x64 fp32 tile of W (row stride 256 floats) into LDS, writing
// LDS rows with stride 80 floats via the pad feature (pad_interval=5 -> pad
// after every 64 DWORDs; pad_amount=15 -> 16 DWORDs of padding).
// D# layout per CDNA5 ISA (groups 2/3 zero => 2-D).
// This toolchain exposes the 6-arg builtin (extra int32x8 group, zeroed).
// -------------------------------------------------------------------------
__device__ __forceinline__ void tdm_load_b_tile(const float* gsrc,
                                                unsigned int lds_off) {
    const unsigned long long ga = (unsigned long long)(size_t)gsrc;
    u32x4 g0;
    g0[0] = 1u;                                     // count=1 (valid), user, load
    g0[1] = lds_off;                                // lds_addr (bytes)
    g0[2] = (unsigned int)ga;                       // global_addr[31:0]
    g0[3] = (unsigned int)(ga >> 32) | (2u << 30);  // global_addr[56:32] | type=2
    i32x8 g1;
    g1[0] = (2 << 16)        // data_size = 4 bytes
          | (1 << 20)        // pad_enable
          | (5 << 22)        // pad_interval code 5 = 64 DWORDs
          | (15 << 25);      // pad_amount  code 15 = 16 DWORDs  -> stride 80 f32
    g1[1] = HID_DIM << 16;   // tensor_dim0[15:0] = 256
    g1[2] = IN_DIM << 16;    // tensor_dim0[31:16]=0 | tensor_dim1[15:0] = 256
    g1[3] = 64 << 16;        // tensor_dim1[31:16]=0 | tile_dim0 = 64
    g1[4] = 64;              // tile_dim1 = 64 rows | tile_dim2 = 0
    g1[5] = HID_DIM;         // tensor_dim0_stride[31:0] = 256
    g1[6] = 0;               // stride hi | tensor_dim1_stride lo
    g1[7] = 0;
    const i32x4 gz4 = {0, 0, 0, 0};
    const i32x8 gz8 = {0, 0, 0, 0, 0, 0, 0, 0};
    __builtin_amdgcn_tensor_load_to_lds(g0, g1, gz4, gz4, gz8, 0);
}

// -------------------------------------------------------------------------
// Kernel 1: h = (x * mask1) @ W  via V_WMMA_F32_16X16X4_F32 (fp32 exact).
// Block = 128 threads = 4 wave32; tile 32(M) x 64(N), K tiled by 64.
// B tiles arrive by TDM (double buffered, pipelined one tile ahead);
// A tiles staged manually to fuse the dropout multiply.
// -------------------------------------------------------------------------
__global__ __launch_bounds__(128)
void gcn_gemm_wmma_f32(const float* __restrict__ x,
                       const float* __restrict__ mask1,
                       const float* __restrict__ W,
                       float* __restrict__ h) {
    __shared__ float As[32][68];        //  8,704 B (stride 68: conflict-free)
    __shared__ float Bs[2][64][80];     // 40,960 B (stride 80 via TDM pad)
    const int tid  = threadIdx.x;
    const int lane = tid & 31;
    const int lm   = lane & 15;         // M (A/C) or N (B) index
    const int kh   = (lane >> 4) << 1;  // K half-select: 0 or 2
    const int wv   = tid >> 5;
    const int mloc = (wv & 1) << 4;     // 0 / 16
    const int nloc = (wv >> 1) << 5;    // 0 / 32
    const int mBase = blockIdx.y * 32;
    const int nBase = blockIdx.x * 64;

    v8f c0 = {};
    v8f c1 = {};

    // Prologue: kick off TDM for the first B tile.
    if (wv == 0)
        tdm_load_b_tile(W + nBase, (unsigned int)(size_t)&Bs[0][0][0]);

    #pragma unroll
    for (int t = 0; t < 4; ++t) {
        const int kt = t * 64;
        // Pipeline: issue TDM for tile t+1 into the other buffer. The barrier
        // at the end of iteration t-1 guarantees that buffer is drained.
        if (wv == 0 && t < 3)
            tdm_load_b_tile(W + (kt + 64) * HID_DIM + nBase,
                            (unsigned int)(size_t)&Bs[(t + 1) & 1][0][0]);

        // ---- stage A tile (32 x 64): fused dropout x*mask1 ----
        for (int i = tid; i < 32 * 16; i += 128) {
            const int r  = i >> 4;
            const int c4 = (i & 15) << 2;
            const float4 xv = *(const float4*)&x    [(mBase + r) * IN_DIM + kt + c4];
            const float4 mv = *(const float4*)&mask1[(mBase + r) * IN_DIM + kt + c4];
            As[r][c4 + 0] = xv.x * mv.x;
            As[r][c4 + 1] = xv.y * mv.y;
            As[r][c4 + 2] = xv.z * mv.z;
            As[r][c4 + 3] = xv.w * mv.w;
        }

        // Tile t complete (<=1 outstanding keeps tile t+1 in flight).
        if (wv == 0) {
            if (t < 3) __builtin_amdgcn_s_wait_tensorcnt(1);
            else       __builtin_amdgcn_s_wait_tensorcnt(0);
        }
        __syncthreads();

        const float (*B)[80] = Bs[t & 1];
        #pragma unroll 4
        for (int k0 = 0; k0 < 64; k0 += 4) {
            // A 16x4 fragment: lane L -> row lm, K = k0 + kh + {0,1}
            const v2f a = *(const v2f*)&As[mloc + lm][k0 + kh];
            // B 4x16 fragments: lane L -> col n, K = k0 + kh + {0,1}
            v2f b0, b1;
            b0.x = B[k0 + kh    ][nloc + lm];
            b0.y = B[k0 + kh + 1][nloc + lm];
            b1.x = B[k0 + kh    ][nloc + 16 + lm];
            b1.y = B[k0 + kh + 1][nloc + 16 + lm];
            c0 = __builtin_amdgcn_wmma_f32_16x16x4_f32(false, a, false, b0,
                                                       (short)0, c0, false, false);
            c1 = __builtin_amdgcn_wmma_f32_16x16x4_f32(false, a, false, b1,
                                                       (short)0, c1, false, false);
        }
        __syncthreads();
    }

    // C/D layout: VGPR v -> M = v + (lane>=16 ? 8 : 0), N = lane%16
    const int grow = mBase + mloc + ((lane >> 4) << 3);
    const int gcol = nBase + nloc + lm;
    #pragma unroll
    for (int v = 0; v < 8; ++v) {
        h[(grow + v) * HID_DIM + gcol     ] = c0[v];
        h[(grow + v) * HID_DIM + gcol + 16] = c1[v];
    }
}

// -------------------------------------------------------------------------
// Kernel 2: out[dst] += edge_val[e] * h[src].  64 lanes per edge, each lane
// gathers a float4 of h (L2-resident: 102 MB < 192 MB L2) and issues 4
// hardware fp32 atomics (global_atomic_add_f32).
// -------------------------------------------------------------------------
__global__ __launch_bounds__(256)
void gcn_spmm_atomic(const int* __restrict__ ei,
                     const float* __restrict__ ev,
                     const float* __restrict__ h,
                     float* __restrict__ out) {
    const long long gid = (long long)blockIdx.x * 256 + threadIdx.x;
    const int e  = (int)(gid >> 6);
    const int c4 = (int)(gid & 63) << 2;
    const int   src = ei[e];
    const int   dst = ei[N_EDGES + e];
    const float a   = ev[e];
    const float4 hv = *(const float4*)&h[(long long)src * HID_DIM + c4];
    float* o = &out[(long long)dst * HID_DIM + c4];
    unsafeAtomicAdd(o + 0, a * hv.x);
    unsafeAtomicAdd(o + 1, a * hv.y);
    unsafeAtomicAdd(o + 2, a * hv.z);
    unsafeAtomicAdd(o + 3, a * hv.w);
}

// -------------------------------------------------------------------------
// Kernel 3: in-place epilogue out = relu(out + bias) * mask2, float4-wide.
// -------------------------------------------------------------------------
__global__ __launch_bounds__(256)
void gcn_epilogue(float* __restrict__ out,
                  const float* __restrict__ bias,
                  const float* __restrict__ mask2) {
    const long long i4 = ((long long)blockIdx.x * 256 + threadIdx.x) << 2;
    const int col = (int)(i4 & (HID_DIM - 1));
    float4 o       = *(float4*)&out[i4];
    const float4 b = *(const float4*)&bias[col];
    const float4 m = *(const float4*)&mask2[i4];
    o.x = fmaxf(o.x + b.x, 0.0f) * m.x;
    o.y = fmaxf(o.y + b.y, 0.0f) * m.y;
    o.z = fmaxf(o.z + b.z, 0.0f) * m.z;
    o.w = fmaxf(o.w + b.w, 0.0f) * m.w;
    *(float4*)&out[i4] = o;
}

extern "C" void kernel_launch(void* const* d_in, const int* in_sizes, int n_in,
                              void* d_out, int out_size, void* d_ws, size_t ws_size,
                              hipStream_t stream) {
    const float* x          = (const float*)d_in[0];
    const int*   edge_index = (const int*)  d_in[1];   // [2, E]: row0=src, row1=dst
    const float* edge_val   = (const float*)d_in[2];
    const float* weight     = (const float*)d_in[3];
    const float* bias       = (const float*)d_in[4];
    const float* mask1      = (const float*)d_in[5];
    const float* mask2      = (const float*)d_in[6];
    float* out = (float*)d_out;
    float* h   = (float*)d_ws;  // needs N_NODES*HID_DIM*4 = 102.4 MB scratch

    (void)in_sizes; (void)n_in; (void)out_size; (void)ws_size;

    // Zero the atomic accumulator every call.
    (void)hipMemsetAsync(out, 0, sizeof(float) * (size_t)N_NODES * HID_DIM, stream);

    // 1) h = (x*mask1) @ W    -- WMMA fp32 + TDM B-tiles
    gcn_gemm_wmma_f32<<<dim3(HID_DIM / 64, N_NODES / 32), 128, 0, stream>>>(
        x, mask1, weight, h);

    // 2) scatter-add messages -- 64 lanes/edge -> exactly 800000 blocks
    gcn_spmm_atomic<<<dim3((N_EDGES * 64) / 256), 256, 0, stream>>>(
        edge_index, edge_val, h, out);

    // 3) bias + ReLU + dropout2, in place -- exactly 25000 blocks
    gcn_epilogue<<<dim3((N_NODES * HID_DIM / 4) / 256), 256, 0, stream>>>(
        out, bias, mask2);
}